// CTCLayer_1254130450965
// MI455X (gfx1250) — compile-verified
//
#include <hip/hip_runtime.h>

// CTC batch cost (keras ctc_batch_cost semantics) for B=256, T=512, C=512, L=64.
// One block per batch row; alpha recursion in LDS; y_pred class rows streamed
// into LDS with gfx1250 async global->LDS copies, 8-deep pipelined, retired
// with s_wait_asynccnt.

#define B_     256
#define T_     512
#define C_     512
#define L_     64
#define S_     129          // 2L+1
#define BLANK_ 511
#define NEG_   (-1e30f)
#define EPS_   (1e-7f)
#define DEPTH  8            // async prefetch depth (rows of 2KB each in LDS)

// --- gfx1250 async global->LDS copy, 16B per lane -------------------------
__device__ __forceinline__ void async_copy_b128(const float* gsrc, float* ldst) {
  // Low 32 bits of a flat pointer into the LDS aperture ARE the LDS offset.
  unsigned lds = (unsigned)(unsigned long long)(void*)ldst;
  unsigned long long ga = (unsigned long long)gsrc;
  asm volatile("global_load_async_to_lds_b128 %0, %1, off"
               :: "v"(lds), "v"(ga)
               : "memory");
}
__device__ __forceinline__ void wait_async_hi() {   // allow DEPTH-2 in flight
  asm volatile("s_wait_asynccnt 6" ::: "memory");
}
__device__ __forceinline__ void wait_async_0() {
  asm volatile("s_wait_asynccnt 0" ::: "memory");
}

// --- 3-way log-sum-exp, matches reference formula -------------------------
__device__ __forceinline__ float lse3(float a1, float a2, float a3) {
  float m = fmaxf(a1, fmaxf(a2, a3));
  return m + __logf(__expf(a1 - m) + __expf(a2 - m) + __expf(a3 - m));
}

__global__ __launch_bounds__(128)
void ctc_loss_kernel(const int* __restrict__ y_true,
                     const float* __restrict__ y_pred,
                     float* __restrict__ out) {
  __shared__ __align__(16) float rowbuf[DEPTH][C_];   // 16 KB: staged class rows
  __shared__ float alpha[2][S_ + 3];                  // ping-pong alpha

  const int tid = threadIdx.x;           // state s == tid; tid 0 also does s=128
  const int b   = blockIdx.x;
  const float* yp  = y_pred + (size_t)b * T_ * C_;
  const int*   lab = y_true + b * L_;

  // Extended-label class + can_skip flag, kept in registers.
  // ext[s] = blank for even s, label[s/2] for odd s.
  // can_skip[s] = odd(s) && s>=3 && label[(s-1)/2] != label[(s-3)/2]
  int myCls, mySkip;
  if (tid & 1) {
    int li = tid >> 1;
    int c  = lab[li];
    myCls  = c;
    mySkip = (tid >= 3) ? (c != lab[li - 1]) : 0;
  } else {
    myCls  = BLANK_;
    mySkip = 0;
  }

  // alpha at t = 0 (direct global loads; only two finite entries)
  if (tid == 0) {
    alpha[0][0]      = __logf(yp[BLANK_] + EPS_);
    alpha[0][S_ - 1] = NEG_;
  } else if (tid == 1) {
    alpha[0][1] = __logf(yp[myCls] + EPS_);
  }
  if (tid >= 2) alpha[0][tid] = NEG_;

  // Prologue: async-stage rows 1..DEPTH-1 (each thread moves 16B => 2KB/row).
  for (int t = 1; t <= DEPTH - 1; ++t)
    async_copy_b128(yp + (size_t)t * C_ + tid * 4,
                    &rowbuf[t & (DEPTH - 1)][tid * 4]);
  wait_async_hi();       // this wave's quarter of row 1 resident
  __syncthreads();       // all quarters + alpha0 visible block-wide

  float* aP = alpha[0];
  float* aN = alpha[1];

  for (int t = 1; t < T_; ++t) {
    // Prefetch row t+DEPTH-1 into the slot freed by row t-1 (consumed last iter).
    const int tpre = t + DEPTH - 1;
    if (tpre < T_)
      async_copy_b128(yp + (size_t)tpre * C_ + tid * 4,
                      &rowbuf[tpre & (DEPTH - 1)][tid * 4]);

    const float* row = rowbuf[t & (DEPTH - 1)];   // guaranteed resident

    // alpha update for state tid
    float a1 = aP[tid];
    float a2 = (tid >= 1) ? aP[tid - 1] : NEG_;
    float a3 = mySkip ? aP[tid - 2] : NEG_;
    float v  = lse3(a1, a2, a3) + __logf(row[myCls] + EPS_);

    float v128 = 0.0f;
    if (tid == 0) {   // state 128: blank, no skip
      float b1 = aP[S_ - 1];
      float b2 = aP[S_ - 2];
      v128 = lse3(b1, b2, NEG_) + __logf(row[BLANK_] + EPS_);
    }

    aN[tid] = v;
    if (tid == 0) aN[S_ - 1] = v128;

    // Retire row t+1 (in-order per-wave async completion), then block sync:
    // covers alpha ping-pong visibility AND cross-wave row quarters.
    if (tpre < T_) wait_async_hi();
    else           wait_async_0();
    __syncthreads();

    float* tmp = aP; aP = aN; aN = tmp;
  }

  // tail = logsumexp(alpha[S-2], alpha[S-1]); loss = -ll
  if (tid == 0) {
    float b1 = aP[S_ - 2];
    float b2 = aP[S_ - 1];
    float m  = fmaxf(b1, b2);
    float ll = m + __logf(__expf(b1 - m) + __expf(b2 - m));
    out[b] = -ll;
  }
}

extern "C" void kernel_launch(void* const* d_in, const int* in_sizes, int n_in,
                              void* d_out, int out_size, void* d_ws, size_t ws_size,
                              hipStream_t stream) {
  (void)in_sizes; (void)n_in; (void)d_ws; (void)ws_size; (void)out_size;
  const int*   y_true = (const int*)d_in[0];    // [256, 64] int32
  const float* y_pred = (const float*)d_in[1];  // [256, 512, 512] f32 softmax
  float*       out    = (float*)d_out;          // [256] f32
  ctc_loss_kernel<<<B_, 128, 0, stream>>>(y_true, y_pred, out);
}